// SSLGNN_42958262894917
// MI455X (gfx1250) — compile-verified
//
#include <hip/hip_runtime.h>
#include <hip/hip_bf16.h>

typedef __attribute__((ext_vector_type(2))) float v2f;
typedef __attribute__((ext_vector_type(8))) float v8f;

#define N_NODES 50000
#define N_EDGES 800000
#define N_GRAPHS 512
#define HID 128

__device__ __forceinline__ v8f wmma_f32(v2f a, v2f b, v8f c) {
    // D = A(16x4) x B(4x16) + C(16x16), fp32 WMMA
    return __builtin_amdgcn_wmma_f32_16x16x4_f32(false, a, false, b, (short)0, c, false, false);
}

// ---------------- node encoder: h = x[50000,32] @ Wn[32,128] + bnode ----------------
__global__ __launch_bounds__(256, 2) void k_node_enc(const float* __restrict__ x,
                                                     const float* __restrict__ Wn,
                                                     const float* __restrict__ bn,
                                                     float* __restrict__ h) {
    const int lane = threadIdx.x & 31;
    const int m = lane & 15;
    const int kb = (lane >> 4) * 2;   // K offset within frag for this half-wave
    const int mb = (lane >> 4) * 8;   // M offset for C layout
    int wave = blockIdx.x * (blockDim.x >> 5) + (threadIdx.x >> 5);
    int nw = gridDim.x * (blockDim.x >> 5);
    for (int rt = wave; rt < N_NODES / 16; rt += nw) {
        const int r0 = rt * 16;
        // hoist all A fragments for this row tile (K=32 -> 8 frags)
        v2f afr[8];
#pragma unroll
        for (int j = 0; j < 8; ++j) {
            const float2 av = *reinterpret_cast<const float2*>(x + (r0 + m) * 32 + 4 * j + kb);
            afr[j].x = av.x; afr[j].y = av.y;
        }
#pragma unroll
        for (int ct = 0; ct < 8; ++ct) {
            const int n0 = ct * 16;
            const float bias = bn[n0 + m];
            v8f acc;
#pragma unroll
            for (int g = 0; g < 8; ++g) acc[g] = bias;
#pragma unroll
            for (int j = 0; j < 8; ++j) {
                const int k0 = 4 * j;
                v2f b;
                b.x = Wn[(k0 + kb) * HID + n0 + m];
                b.y = Wn[(k0 + kb + 1) * HID + n0 + m];
                acc = wmma_f32(afr[j], b, acc);
            }
#pragma unroll
            for (int g = 0; g < 8; ++g) h[(r0 + mb + g) * HID + n0 + m] = acc[g];
        }
    }
}

// ---- fused edge-encoder + message + scatter-add:
//      aggr[dst] += relu(h[src] + edge_attr@We + bedge)   (recompute e -> 8x less HBM traffic)
__global__ __launch_bounds__(256, 4) void k_msg_scatter(const float* __restrict__ h,
                                                        const float* __restrict__ eattr,
                                                        const int* __restrict__ eidx,
                                                        const float* __restrict__ We,
                                                        const float* __restrict__ be,
                                                        float* __restrict__ aggr) {
    __shared__ float4 sWe4[512];   // We[16][128] as float4 columns
    __shared__ float4 sbe4[32];
    const float4* We4 = (const float4*)We;
    for (int i = threadIdx.x; i < 512; i += blockDim.x) sWe4[i] = We4[i];
    if (threadIdx.x < 32) sbe4[threadIdx.x] = ((const float4*)be)[threadIdx.x];
    __syncthreads();

    const int lane = threadIdx.x & 31;
    int wave = blockIdx.x * (blockDim.x >> 5) + (threadIdx.x >> 5);
    int nw = gridDim.x * (blockDim.x >> 5);
    const float4* h4 = (const float4*)h;
    for (int ed = wave; ed < N_EDGES; ed += nw) {
        const int s = eidx[ed];
        const int d = eidx[N_EDGES + ed];
        const float4* ap = (const float4*)(eattr + ed * 16);
        const float4 a0 = ap[0], a1 = ap[1], a2 = ap[2], a3 = ap[3];
        const float ak[16] = {a0.x, a0.y, a0.z, a0.w, a1.x, a1.y, a1.z, a1.w,
                              a2.x, a2.y, a2.z, a2.w, a3.x, a3.y, a3.z, a3.w};
        float4 acc = sbe4[lane];
#pragma unroll
        for (int k = 0; k < 16; ++k) {
            const float4 wv = sWe4[k * 32 + lane];
            acc.x = fmaf(ak[k], wv.x, acc.x);
            acc.y = fmaf(ak[k], wv.y, acc.y);
            acc.z = fmaf(ak[k], wv.z, acc.z);
            acc.w = fmaf(ak[k], wv.w, acc.w);
        }
        const float4 hv = h4[s * 32 + lane];
        const float mx = fmaxf(hv.x + acc.x, 0.f);
        const float my = fmaxf(hv.y + acc.y, 0.f);
        const float mz = fmaxf(hv.z + acc.z, 0.f);
        const float mw = fmaxf(hv.w + acc.w, 0.f);
        float* dp = aggr + d * HID + lane * 4;
        atomicAdd(dp + 0, mx);
        atomicAdd(dp + 1, my);
        atomicAdd(dp + 2, mz);
        atomicAdd(dp + 3, mw);
    }
}

// ---- fused GINE node MLP: v = relu((h+aggr)@W1 + b1)@W2 + b2, plus BN column stats ----
#define MLP_WAVES 6
#define SCR_STRIDE 132   // 16-row tile, padded: bank-conflict-free A-frag reads & C-layout writes
__global__ __launch_bounds__(MLP_WAVES * 32, 1) void k_gine_mlp(
        const float* __restrict__ h, const float* __restrict__ aggr,
        const float* __restrict__ W1, const float* __restrict__ b1,
        const float* __restrict__ W2, const float* __restrict__ b2,
        float* __restrict__ v, float* __restrict__ csum, float* __restrict__ csq) {
    extern __shared__ float smem[];  // MLP_WAVES * 16 * SCR_STRIDE floats
    const int lane = threadIdx.x & 31;
    const int wid = threadIdx.x >> 5;
    float* scr = smem + wid * (16 * SCR_STRIDE);
    const int m = lane & 15;
    const int kb = (lane >> 4) * 2;
    const int mb = (lane >> 4) * 8;
    int wave = blockIdx.x * MLP_WAVES + wid;
    int nw = gridDim.x * MLP_WAVES;
    const float4* h4 = (const float4*)h;
    const float4* a4 = (const float4*)aggr;

    for (int rt = wave; rt < N_NODES / 16; rt += nw) {
        const int r0 = rt * 16;
        // stage A = h + aggr into per-wave LDS scratch (row-major, padded)
#pragma unroll
        for (int r = 0; r < 16; ++r) {
            const float4 hv = h4[(r0 + r) * 32 + lane];
            const float4 av = a4[(r0 + r) * 32 + lane];
            float4 u;
            u.x = hv.x + av.x; u.y = hv.y + av.y; u.z = hv.z + av.z; u.w = hv.w + av.w;
            *reinterpret_cast<float4*>(scr + r * SCR_STRIDE + lane * 4) = u;
        }
        // ---- GEMM1: t = relu(A @ W1 + b1) ----
        v8f acc[8];
#pragma unroll
        for (int ct = 0; ct < 8; ++ct) {
            const float bb = b1[ct * 16 + m];
#pragma unroll
            for (int g = 0; g < 8; ++g) acc[ct][g] = bb;
        }
        for (int k0 = 0; k0 < HID; k0 += 4) {
            v2f a;
            const float2 av = *reinterpret_cast<const float2*>(scr + m * SCR_STRIDE + k0 + kb);
            a.x = av.x; a.y = av.y;
#pragma unroll
            for (int ct = 0; ct < 8; ++ct) {
                v2f b;
                b.x = W1[(k0 + kb) * HID + ct * 16 + m];
                b.y = W1[(k0 + kb + 1) * HID + ct * 16 + m];
                acc[ct] = wmma_f32(a, b, acc[ct]);
            }
        }
        // relu + transpose C-layout -> row-major LDS (becomes A of GEMM2)
#pragma unroll
        for (int ct = 0; ct < 8; ++ct)
#pragma unroll
            for (int g = 0; g < 8; ++g)
                scr[(mb + g) * SCR_STRIDE + ct * 16 + m] = fmaxf(acc[ct][g], 0.f);
        // ---- GEMM2: v = t @ W2 + b2 ----
#pragma unroll
        for (int ct = 0; ct < 8; ++ct) {
            const float bb = b2[ct * 16 + m];
#pragma unroll
            for (int g = 0; g < 8; ++g) acc[ct][g] = bb;
        }
        for (int k0 = 0; k0 < HID; k0 += 4) {
            v2f a;
            const float2 av = *reinterpret_cast<const float2*>(scr + m * SCR_STRIDE + k0 + kb);
            a.x = av.x; a.y = av.y;
#pragma unroll
            for (int ct = 0; ct < 8; ++ct) {
                v2f b;
                b.x = W2[(k0 + kb) * HID + ct * 16 + m];
                b.y = W2[(k0 + kb + 1) * HID + ct * 16 + m];
                acc[ct] = wmma_f32(a, b, acc[ct]);
            }
        }
        // store v + accumulate BN column stats (half-wave reduce via shfl_xor(16))
#pragma unroll
        for (int ct = 0; ct < 8; ++ct) {
            float s = 0.f, q = 0.f;
#pragma unroll
            for (int g = 0; g < 8; ++g) {
                const float t = acc[ct][g];
                v[(r0 + mb + g) * HID + ct * 16 + m] = t;
                s += t;
                q += t * t;
            }
            s += __shfl_xor(s, 16, 32);
            q += __shfl_xor(q, 16, 32);
            if (lane < 16) {
                atomicAdd(&csum[ct * 16 + lane], s);
                atomicAdd(&csq[ct * 16 + lane], q);
            }
        }
    }
}

// ---- BN (training stats, biased var) + relu ----
__global__ __launch_bounds__(256, 4) void k_bn_relu(const float* __restrict__ vin,
                                                    const float* __restrict__ csum,
                                                    const float* __restrict__ csq,
                                                    const float* __restrict__ gamma,
                                                    const float* __restrict__ beta,
                                                    float* __restrict__ out,
                                                    int total, float invM) {
    int i = blockIdx.x * blockDim.x + threadIdx.x;
    const int stride = gridDim.x * blockDim.x;
    for (; i < total; i += stride) {
        const int c = i & (HID - 1);
        const float mu = csum[c] * invM;
        const float var = csq[c] * invM - mu * mu;
        const float sc = gamma[c] * rsqrtf(var + 1e-5f);
        out[i] = fmaxf((vin[i] - mu) * sc + beta[c], 0.f);
    }
}

// ---- global mean pool (atomics into 512x128, counts) ----
__global__ __launch_bounds__(256, 4) void k_pool(const float* __restrict__ h,
                                                 const int* __restrict__ batch,
                                                 float* __restrict__ gp, float* __restrict__ cnt) {
    const int lane = threadIdx.x & 31;
    int wave = blockIdx.x * (blockDim.x >> 5) + (threadIdx.x >> 5);
    int nw = gridDim.x * (blockDim.x >> 5);
    const float4* h4 = (const float4*)h;
    for (int i = wave; i < N_NODES; i += nw) {
        const int b = batch[i];
        const float4 hv = h4[i * 32 + lane];
        float* p = gp + b * HID + lane * 4;
        atomicAdd(p + 0, hv.x);
        atomicAdd(p + 1, hv.y);
        atomicAdd(p + 2, hv.z);
        atomicAdd(p + 3, hv.w);
        if (lane == 0) atomicAdd(&cnt[b], 1.0f);
    }
}

__global__ void k_pool_div(float* __restrict__ gp, const float* __restrict__ cnt) {
    const int i = blockIdx.x * blockDim.x + threadIdx.x;
    if (i < N_GRAPHS * HID) gp[i] /= fmaxf(cnt[i >> 7], 1.0f);
}

// ---- generic small WMMA GEMM: out = A[M,K]@W[K,N] + bias, optional BN column stats ----
__global__ __launch_bounds__(256, 2) void k_gemm_stats(const float* __restrict__ A,
                                                       const float* __restrict__ W,
                                                       const float* __restrict__ bias,
                                                       float* __restrict__ out,
                                                       int M, int K, int N,
                                                       float* __restrict__ csum,
                                                       float* __restrict__ csq, int doStats) {
    const int lane = threadIdx.x & 31;
    const int m = lane & 15;
    const int kb = (lane >> 4) * 2;
    const int mb = (lane >> 4) * 8;
    int wave = blockIdx.x * (blockDim.x >> 5) + (threadIdx.x >> 5);
    int nw = gridDim.x * (blockDim.x >> 5);
    const int ntile = N >> 4;
    const int tiles = (M >> 4) * ntile;
    for (int t = wave; t < tiles; t += nw) {
        const int rt = t / ntile, ct = t - rt * ntile;
        const int r0 = rt * 16, n0 = ct * 16;
        v8f acc;
        const float bb = bias[n0 + m];
#pragma unroll
        for (int g = 0; g < 8; ++g) acc[g] = bb;
        for (int k0 = 0; k0 < K; k0 += 4) {
            v2f a, b;
            const float2 av = *reinterpret_cast<const float2*>(A + (r0 + m) * K + k0 + kb);
            a.x = av.x; a.y = av.y;
            b.x = W[(k0 + kb) * N + n0 + m];
            b.y = W[(k0 + kb + 1) * N + n0 + m];
            acc = wmma_f32(a, b, acc);
        }
        float s = 0.f, q = 0.f;
#pragma unroll
        for (int g = 0; g < 8; ++g) {
            const float t2 = acc[g];
            out[(r0 + mb + g) * N + n0 + m] = t2;
            s += t2;
            q += t2 * t2;
        }
        if (doStats) {
            s += __shfl_xor(s, 16, 32);
            q += __shfl_xor(q, 16, 32);
            if (lane < 16) {
                atomicAdd(&csum[n0 + lane], s);
                atomicAdd(&csq[n0 + lane], q);
            }
        }
    }
}

extern "C" void kernel_launch(void* const* d_in, const int* in_sizes, int n_in,
                              void* d_out, int out_size, void* d_ws, size_t ws_size,
                              hipStream_t stream) {
    const float* x      = (const float*)d_in[0];
    const float* eattr  = (const float*)d_in[1];
    const int*   eidx   = (const int*)d_in[2];
    const int*   batch  = (const int*)d_in[3];
    const float* Wn     = (const float*)d_in[4];
    const float* bnode  = (const float*)d_in[5];
    const float* We     = (const float*)d_in[6];
    const float* bedge  = (const float*)d_in[7];
    const float* convW1 = (const float*)d_in[8];
    const float* convb1 = (const float*)d_in[9];
    const float* convW2 = (const float*)d_in[10];
    const float* convb2 = (const float*)d_in[11];
    const float* bng    = (const float*)d_in[12];
    const float* bnb    = (const float*)d_in[13];
    const float* embW1  = (const float*)d_in[14];
    const float* embb1  = (const float*)d_in[15];
    const float* embg   = (const float*)d_in[16];
    const float* embbt  = (const float*)d_in[17];
    const float* embW2  = (const float*)d_in[18];
    const float* embb2  = (const float*)d_in[19];
    const float* prjW1  = (const float*)d_in[20];
    const float* prjb1  = (const float*)d_in[21];
    const float* prjg   = (const float*)d_in[22];
    const float* prjbt  = (const float*)d_in[23];
    const float* prjW2  = (const float*)d_in[24];
    const float* prjb2  = (const float*)d_in[25];

    float* ws   = (float*)d_ws;
    float* h    = ws;                    // 50000*128
    float* aggr = h + 6400000;           // 50000*128
    float* vv   = aggr + 6400000;        // 50000*128
    float* cs   = vv + 6400000;          // 128
    float* cq   = cs + 128;              // 128
    float* gp   = cq + 128;              // 512*128
    float* cnt  = gp + 65536;            // 512
    float* t1   = cnt + 512;             // 512*128
    float* t2   = t1 + 65536;            // 512*128
    float* zout = (float*)d_out;         // 512*128
    float* pout = zout + 512 * 128;      // 512*64

    const size_t mlp_lds = (size_t)MLP_WAVES * 16 * SCR_STRIDE * sizeof(float); // ~50.7 KB

    k_node_enc<<<391, 256, 0, stream>>>(x, Wn, bnode, h);
    for (int l = 0; l < 3; ++l) {
        hipMemsetAsync(aggr, 0, 6400000 * sizeof(float), stream);
        hipMemsetAsync(cs, 0, 256 * sizeof(float), stream);
        k_msg_scatter<<<1024, 256, 0, stream>>>(h, eattr, eidx, We, bedge, aggr);
        k_gine_mlp<<<200, MLP_WAVES * 32, mlp_lds, stream>>>(
            h, aggr, convW1 + l * 16384, convb1 + l * 128,
            convW2 + l * 16384, convb2 + l * 128, vv, cs, cq);
        k_bn_relu<<<4096, 256, 0, stream>>>(vv, cs, cq, bng + l * 128, bnb + l * 128, h,
                                            6400000, 1.0f / 50000.0f);
    }
    hipMemsetAsync(gp, 0, (65536 + 512) * sizeof(float), stream);
    k_pool<<<1024, 256, 0, stream>>>(h, batch, gp, cnt);
    k_pool_div<<<256, 256, 0, stream>>>(gp, cnt);

    // embedding head
    hipMemsetAsync(cs, 0, 256 * sizeof(float), stream);
    k_gemm_stats<<<32, 256, 0, stream>>>(gp, embW1, embb1, t1, 512, 128, 128, cs, cq, 1);
    k_bn_relu<<<128, 256, 0, stream>>>(t1, cs, cq, embg, embbt, t1, 65536, 1.0f / 512.0f);
    k_gemm_stats<<<32, 256, 0, stream>>>(t1, embW2, embb2, zout, 512, 128, 128, nullptr, nullptr, 0);

    // projection head
    hipMemsetAsync(cs, 0, 256 * sizeof(float), stream);
    k_gemm_stats<<<32, 256, 0, stream>>>(zout, prjW1, prjb1, t2, 512, 128, 128, cs, cq, 1);
    k_bn_relu<<<128, 256, 0, stream>>>(t2, cs, cq, prjg, prjbt, t2, 65536, 1.0f / 512.0f);
    k_gemm_stats<<<16, 256, 0, stream>>>(t2, prjW2, prjb2, pout, 512, 128, 64, nullptr, nullptr, 0);
}